// KalmanFilter_48919677501725
// MI455X (gfx1250) — compile-verified
//
#include <hip/hip_runtime.h>
#include <hip/hip_bf16.h>

// Batched Kalman filter + RTS smoother for MI455X (gfx1250, wave32).
// One wave per batch element; all 16x16 matmuls via V_WMMA_F32_16X16X4_F32
// (4 chained WMMAs = one fp32 16x16x16 matmul, full reference precision).
// Sequential T=256 scan => latency-bound: everything stays in LDS/VGPRs,
// Cholesky solves are fully unrolled and done redundantly per lane.

typedef __attribute__((ext_vector_type(2))) float v2f;
typedef __attribute__((ext_vector_type(8))) float v8f;

#define TT     256
#define BATCH  256
#define NWAVES 8
#define QVAR   0.01f   // STD_DYN^2
#define RVAR   0.01f   // STD_OBS^2

__device__ __forceinline__ v8f vzero8() {
  v8f z;
#pragma unroll
  for (int i = 0; i < 8; ++i) z[i] = 0.0f;
  return z;
}

__device__ __forceinline__ v8f wmma4(v2f a, v2f b, v8f c) {
  // D(16x16,f32) = A(16x4,f32) x B(4x16,f32) + C
  return __builtin_amdgcn_wmma_f32_16x16x4_f32(false, a, false, b, (short)0, c,
                                               false, false);
}

// Full 16x16x16 fp32 matmul: D += X * W  (optionally X^T-as-A / W^T-as-B reads)
// X, W are LDS row-major 16x16 (stride 16).
template <bool TA, bool TB>
__device__ __forceinline__ v8f mm16x16(const float* X, const float* W, v8f c,
                                       int col, int hi) {
#pragma unroll
  for (int k = 0; k < 4; ++k) {
    const int ka = 4 * k + 2 * hi;  // A-operand K pair base
    const int kb = 4 * k + hi;      // B-operand K per VGPR
    v2f a, b;
    if (TA) { a[0] = X[ka * 16 + col];      a[1] = X[(ka + 1) * 16 + col]; }
    else    { a[0] = X[col * 16 + ka];      a[1] = X[col * 16 + ka + 1]; }
    if (TB) { b[0] = W[col * 16 + kb];      b[1] = W[col * 16 + kb + 2]; }
    else    { b[0] = W[kb * 16 + col];      b[1] = W[(kb + 2) * 16 + col]; }
    c = wmma4(a, b, c);
  }
  return c;
}

__global__ __launch_bounds__(256, 1) void kalman_rts_kernel(
    const float* __restrict__ Y, const float* __restrict__ Uc,
    const float* __restrict__ Aseq, const float* __restrict__ Bseq,
    const float* __restrict__ Cseq, const float* __restrict__ mu0,
    const float* __restrict__ Sig0, float* __restrict__ out) {
  // Block-shared per-timestep system matrices (same for all batch elements)
  __shared__ float sA[256];  // A_t 16x16
  __shared__ float sB[64];   // B_t 16x4
  __shared__ float sC[128];  // C_t 8x16
  // Per-wave scratch
  __shared__ float wSig[NWAVES][256];  // covariance carry
  __shared__ float wSP [NWAVES][256];  // Sig_pred (filter) / Sig_p[t+1] (smoother)
  __shared__ float wT  [NWAVES][256];  // temp
  __shared__ float wT2 [NWAVES][256];  // temp
  __shared__ float wICK[NWAVES][256];  // I-KC (filter) / J (smoother)
  __shared__ float wK  [NWAVES][128];  // Kalman gain 16x8
  __shared__ float wPCT[NWAVES][128];  // Sig_pred C^T 16x8
  __shared__ float wS  [NWAVES][72];   // innovation cov 8x8 (stride 9)
  __shared__ float wmu [NWAVES][16];
  __shared__ float wmp [NWAVES][16];   // mu_pred / mu_p[t+1]
  __shared__ float wr  [NWAVES][16];
  __shared__ float wv  [NWAVES][16];

  const int tid  = threadIdx.x;
  const int w    = tid >> 5;
  const int lane = tid & 31;
  const int col  = lane & 15;  // N index of D/C fragments
  const int hi   = lane >> 4;  // lane half
  const int b    = blockIdx.x * NWAVES + w;

  const long O1 = (long)BATCH * TT * 16;        // Sigs_smooth
  const long O2 = O1 + (long)BATCH * TT * 256;  // mus_f
  const long O3 = O2 + (long)BATCH * TT * 16;   // Sigs_f

  // init carry: mu = mu0, Sig = Sigma0
  if (lane < 16) wmu[w][lane] = mu0[lane];
#pragma unroll
  for (int r = 0; r < 8; ++r) {
    const int m = r + 8 * hi;
    wSig[w][m * 16 + col] = Sig0[m * 16 + col];
  }

  // ============================ forward filter ============================
  for (int t = 0; t < TT; ++t) {
    sA[tid] = Aseq[t * 256 + tid];
    if (tid < 64)  sB[tid] = Bseq[t * 64 + tid];
    if (tid < 128) sC[tid] = Cseq[t * 128 + tid];
    if (t + 1 < TT) __builtin_prefetch(&Aseq[(t + 1) * 256 + tid], 0, 1);
    __syncthreads();

    // mu_pred = A mu + B u   (also stash into mus_smooth slot as scratch)
    if (lane < 16) {
      float mp = 0.f;
#pragma unroll
      for (int j = 0; j < 16; ++j) mp += sA[lane * 16 + j] * wmu[w][j];
#pragma unroll
      for (int j = 0; j < 4; ++j) mp += sB[lane * 4 + j] * Uc[((long)b * TT + t) * 4 + j];
      wmp[w][lane] = mp;
      out[((long)b * TT + t) * 16 + lane] = mp;
    }
    // T1 = Sig * A^T
    v8f c1 = vzero8();
    c1 = mm16x16<false, true>(wSig[w], sA, c1, col, hi);
#pragma unroll
    for (int r = 0; r < 8; ++r) wT[w][(r + 8 * hi) * 16 + col] = c1[r];
    __syncthreads();

    // Sig_pred = A * T1 + Q  -> wSP and Sigs_smooth slot (scratch for smoother)
    v8f c2 = vzero8();
    c2 = mm16x16<false, false>(sA, wT[w], c2, col, hi);
    const long spg = O1 + ((long)b * TT + t) * 256;
#pragma unroll
    for (int r = 0; r < 8; ++r) {
      const int m = r + 8 * hi;
      const float v = c2[r] + ((m == col) ? QVAR : 0.f);
      wSP[w][m * 16 + col] = v;
      out[spg + m * 16 + col] = v;
    }
    // innovation r = y - C mu_pred
    if (lane < 8) {
      float rr = Y[((long)b * TT + t) * 8 + lane];
#pragma unroll
      for (int j = 0; j < 16; ++j) rr -= sC[lane * 16 + j] * wmp[w][j];
      wr[w][lane] = rr;
    }
    __syncthreads();

    // PCT = Sig_pred * C^T   (B operand = C^T, cols >= 8 zero)
    v8f c3 = vzero8();
#pragma unroll
    for (int k = 0; k < 4; ++k) {
      const int ka = 4 * k + 2 * hi, kb = 4 * k + hi;
      v2f a, bb;
      a[0] = wSP[w][col * 16 + ka]; a[1] = wSP[w][col * 16 + ka + 1];
      bb[0] = (col < 8) ? sC[col * 16 + kb] : 0.f;
      bb[1] = (col < 8) ? sC[col * 16 + kb + 2] : 0.f;
      c3 = wmma4(a, bb, c3);
    }
    if (col < 8) {
#pragma unroll
      for (int r = 0; r < 8; ++r) wPCT[w][(r + 8 * hi) * 8 + col] = c3[r];
    }
    __syncthreads();

    // S = C * PCT + R   (valid 8x8 block)
    v8f c4 = vzero8();
#pragma unroll
    for (int k = 0; k < 4; ++k) {
      const int ka = 4 * k + 2 * hi, kb = 4 * k + hi;
      v2f a, bb;
      a[0] = (col < 8) ? sC[col * 16 + ka] : 0.f;
      a[1] = (col < 8) ? sC[col * 16 + ka + 1] : 0.f;
      bb[0] = (col < 8) ? wPCT[w][kb * 8 + col] : 0.f;
      bb[1] = (col < 8) ? wPCT[w][(kb + 2) * 8 + col] : 0.f;
      c4 = wmma4(a, bb, c4);
    }
    if (hi == 0 && col < 8) {
#pragma unroll
      for (int r = 0; r < 8; ++r)
        wS[w][r * 9 + col] = c4[r] + ((r == col) ? RVAR : 0.f);
    }
    __syncthreads();

    // Cholesky 8x8 (redundant per lane), then K rows: S K^T = PCT^T
    {
      float Ld[8][8]; float invd[8];
#pragma unroll
      for (int j = 0; j < 8; ++j) {
        float d = wS[w][j * 9 + j];
#pragma unroll
        for (int k = 0; k < j; ++k) d -= Ld[j][k] * Ld[j][k];
        d = sqrtf(fmaxf(d, 1e-30f));
        Ld[j][j] = d;
        const float id = 1.f / d; invd[j] = id;
#pragma unroll
        for (int i = j + 1; i < 8; ++i) {
          float s = wS[w][i * 9 + j];
#pragma unroll
          for (int k = 0; k < j; ++k) s -= Ld[i][k] * Ld[j][k];
          Ld[i][j] = s * id;
        }
      }
      if (lane < 16) {
        float x[8];
#pragma unroll
        for (int k = 0; k < 8; ++k) x[k] = wPCT[w][lane * 8 + k];
#pragma unroll
        for (int j = 0; j < 8; ++j) {
          float s = x[j];
#pragma unroll
          for (int k = 0; k < j; ++k) s -= Ld[j][k] * x[k];
          x[j] = s * invd[j];
        }
#pragma unroll
        for (int j = 7; j >= 0; --j) {
          float s = x[j];
#pragma unroll
          for (int k = j + 1; k < 8; ++k) s -= Ld[k][j] * x[k];
          x[j] = s * invd[j];
        }
#pragma unroll
        for (int k = 0; k < 8; ++k) wK[w][lane * 8 + k] = x[k];
      }
    }
    __syncthreads();

    // mu_f = mu_pred + K r
    if (lane < 16) {
      float mf = wmp[w][lane];
#pragma unroll
      for (int k = 0; k < 8; ++k) mf += wK[w][lane * 8 + k] * wr[w][k];
      wmu[w][lane] = mf;
      out[O2 + ((long)b * TT + t) * 16 + lane] = mf;
    }
    // I - K C
#pragma unroll
    for (int r = 0; r < 8; ++r) {
      const int m = r + 8 * hi;
      float s = (m == col) ? 1.f : 0.f;
#pragma unroll
      for (int k = 0; k < 8; ++k) s -= wK[w][m * 8 + k] * sC[k * 16 + col];
      wICK[w][m * 16 + col] = s;
    }
    __syncthreads();

    // T3 = Sig_pred * (I-KC)^T
    v8f c5 = vzero8();
    c5 = mm16x16<false, true>(wSP[w], wICK[w], c5, col, hi);
#pragma unroll
    for (int r = 0; r < 8; ++r) wT[w][(r + 8 * hi) * 16 + col] = c5[r];
    __syncthreads();

    // X = (I-KC) * T3 + R * K K^T   (Joseph form)
    v8f c6 = vzero8();
    c6 = mm16x16<false, false>(wICK[w], wT[w], c6, col, hi);
#pragma unroll
    for (int r = 0; r < 8; ++r) {
      const int m = r + 8 * hi;
      float s = 0.f;
#pragma unroll
      for (int k = 0; k < 8; ++k) s += wK[w][m * 8 + k] * wK[w][col * 8 + k];
      wT2[w][m * 16 + col] = c6[r] + RVAR * s;
    }
    __syncthreads();

    // Sig_f = sym(X) -> carry + Sigs_f output
    const long fb = O3 + ((long)b * TT + t) * 256;
#pragma unroll
    for (int r = 0; r < 8; ++r) {
      const int m = r + 8 * hi;
      const float s = 0.5f * (wT2[w][m * 16 + col] + wT2[w][col * 16 + m]);
      wSig[w][m * 16 + col] = s;
      out[fb + m * 16 + col] = s;
    }
    __syncthreads();
  }

  // ============================ RTS smoother =============================
  // Smooth slots currently hold mu_pred / Sig_pred (scratch). Prefetch slot
  // T-1 into the carry buffers, then overwrite it with the filter values.
  {
    const int t1 = TT - 1;
    v8f pp; float mup = 0.f;
#pragma unroll
    for (int r = 0; r < 8; ++r) {
      const int m = r + 8 * hi;
      pp[r] = out[O1 + ((long)b * TT + t1) * 256 + m * 16 + col];
    }
    if (lane < 16) mup = out[((long)b * TT + t1) * 16 + lane];
#pragma unroll
    for (int r = 0; r < 8; ++r) {
      const int m = r + 8 * hi;
      wSP[w][m * 16 + col] = pp[r];
      out[O1 + ((long)b * TT + t1) * 256 + m * 16 + col] = wSig[w][m * 16 + col];
    }
    if (lane < 16) {
      wmp[w][lane] = mup;
      out[((long)b * TT + t1) * 16 + lane] = wmu[w][lane];
    }
  }
  __syncthreads();

  for (int t = TT - 2; t >= 0; --t) {
    sA[tid] = Aseq[(t + 1) * 256 + tid];  // A_{t+1}
    __syncthreads();

    // Sig_f[t], mu_f[t]
    v8f sigf; float muf = 0.f;
    const long fb = O3 + ((long)b * TT + t) * 256;
#pragma unroll
    for (int r = 0; r < 8; ++r) {
      const int m = r + 8 * hi;
      sigf[r] = out[fb + m * 16 + col];
      wT[w][m * 16 + col] = sigf[r];
    }
    if (lane < 16) {
      muf = out[O2 + ((long)b * TT + t) * 16 + lane];
      wv[w][lane] = wmu[w][lane] - wmp[w][lane];  // mu_next - mu_p_next
    }
    __syncthreads();

    // SfAT = Sig_f * A_{t+1}^T
    v8f c1 = vzero8();
    c1 = mm16x16<false, true>(wT[w], sA, c1, col, hi);
#pragma unroll
    for (int r = 0; r < 8; ++r) wT2[w][(r + 8 * hi) * 16 + col] = c1[r];
    __syncthreads();

    // Cholesky 16x16 of Sig_p[t+1] (redundant per lane); J rows solve
    // Sig_p J^T = SfAT^T  (one RHS per lane)
    {
      float Ld[16][16]; float invd[16];
#pragma unroll
      for (int j = 0; j < 16; ++j) {
        float d = wSP[w][j * 16 + j];
#pragma unroll
        for (int k = 0; k < j; ++k) d -= Ld[j][k] * Ld[j][k];
        d = sqrtf(fmaxf(d, 1e-30f));
        Ld[j][j] = d;
        const float id = 1.f / d; invd[j] = id;
#pragma unroll
        for (int i = j + 1; i < 16; ++i) {
          float s = wSP[w][i * 16 + j];
#pragma unroll
          for (int k = 0; k < j; ++k) s -= Ld[i][k] * Ld[j][k];
          Ld[i][j] = s * id;
        }
      }
      if (lane < 16) {
        float x[16];
#pragma unroll
        for (int k = 0; k < 16; ++k) x[k] = wT2[w][lane * 16 + k];
#pragma unroll
        for (int j = 0; j < 16; ++j) {
          float s = x[j];
#pragma unroll
          for (int k = 0; k < j; ++k) s -= Ld[j][k] * x[k];
          x[j] = s * invd[j];
        }
#pragma unroll
        for (int j = 15; j >= 0; --j) {
          float s = x[j];
#pragma unroll
          for (int k = j + 1; k < 16; ++k) s -= Ld[k][j] * x[k];
          x[j] = s * invd[j];
        }
#pragma unroll
        for (int k = 0; k < 16; ++k) wICK[w][lane * 16 + k] = x[k];  // J
      }
    }
    __syncthreads();

    // mu_s = mu_f + J v
    float ms = 0.f;
    if (lane < 16) {
      ms = muf;
#pragma unroll
      for (int k = 0; k < 16; ++k) ms += wICK[w][lane * 16 + k] * wv[w][k];
    }
    // dSig = Sig_next - Sig_p_next
#pragma unroll
    for (int r = 0; r < 8; ++r) {
      const int m = r + 8 * hi;
      wT[w][m * 16 + col] = wSig[w][m * 16 + col] - wSP[w][m * 16 + col];
    }
    __syncthreads();

    // Uu = dSig * J^T
    v8f c2 = vzero8();
    c2 = mm16x16<false, true>(wT[w], wICK[w], c2, col, hi);
#pragma unroll
    for (int r = 0; r < 8; ++r) wT2[w][(r + 8 * hi) * 16 + col] = c2[r];
    __syncthreads();

    // X = J * Uu + Sig_f; prefetch Sig_p[t]/mu_p[t] before overwriting slot t
    v8f c3 = vzero8();
    c3 = mm16x16<false, false>(wICK[w], wT2[w], c3, col, hi);
#pragma unroll
    for (int r = 0; r < 8; ++r) c3[r] += sigf[r];

    v8f pp; float mup = 0.f;
    const long sb = O1 + ((long)b * TT + t) * 256;
#pragma unroll
    for (int r = 0; r < 8; ++r) pp[r] = out[sb + (r + 8 * hi) * 16 + col];
    if (lane < 16) mup = out[((long)b * TT + t) * 16 + lane];
    __syncthreads();

#pragma unroll
    for (int r = 0; r < 8; ++r) {
      const int m = r + 8 * hi;
      wSP[w][m * 16 + col] = pp[r];   // carry Sig_p[t] for next iteration
      wT[w][m * 16 + col] = c3[r];    // stash X for symmetrization
    }
    if (lane < 16) wmp[w][lane] = mup;
    __syncthreads();

    // Sig_s = sym(X) -> carry + smoothed outputs
#pragma unroll
    for (int r = 0; r < 8; ++r) {
      const int m = r + 8 * hi;
      const float s = 0.5f * (wT[w][m * 16 + col] + wT[w][col * 16 + m]);
      wSig[w][m * 16 + col] = s;
      out[sb + m * 16 + col] = s;
    }
    if (lane < 16) {
      wmu[w][lane] = ms;
      out[((long)b * TT + t) * 16 + lane] = ms;
    }
    __syncthreads();
  }
}

extern "C" void kernel_launch(void* const* d_in, const int* in_sizes, int n_in,
                              void* d_out, int out_size, void* d_ws,
                              size_t ws_size, hipStream_t stream) {
  const float* Y    = (const float*)d_in[0];
  const float* Uc   = (const float*)d_in[1];
  const float* Aseq = (const float*)d_in[2];
  const float* Bseq = (const float*)d_in[3];
  const float* Cseq = (const float*)d_in[4];
  const float* mu0  = (const float*)d_in[5];
  const float* Sig0 = (const float*)d_in[6];
  float* out = (float*)d_out;

  dim3 grid(BATCH / NWAVES);
  dim3 block(256);
  hipLaunchKernelGGL(kalman_rts_kernel, grid, block, 0, stream,
                     Y, Uc, Aseq, Bseq, Cseq, mu0, Sig0, out);
}